// Graph_learn_42820823941240
// MI455X (gfx1250) — compile-verified
//
#include <hip/hip_runtime.h>
#include <math.h>

#define NNODES 1000
#define NPAD   1024
#define NB     16
#define DNODE  64
#define DMODEL 128
#define LEPS   1e-5f

typedef float v2f __attribute__((ext_vector_type(2)));
typedef float v8f __attribute__((ext_vector_type(8)));

// ----------------------------- block reductions (wave32) -----------------------------
__device__ __forceinline__ float block_sum(float v, float* sh) {
#pragma unroll
  for (int o = 16; o > 0; o >>= 1) v += __shfl_xor(v, o, 32);
  const int lane = threadIdx.x & 31, w = threadIdx.x >> 5;
  const int nw = (blockDim.x + 31) >> 5;
  __syncthreads();
  if (lane == 0) sh[w] = v;
  __syncthreads();
  if (w == 0) {
    float s = (lane < nw) ? sh[lane] : 0.f;
#pragma unroll
    for (int o = 16; o > 0; o >>= 1) s += __shfl_xor(s, o, 32);
    if (lane == 0) sh[0] = s;
  }
  __syncthreads();
  return sh[0];
}

__device__ __forceinline__ float block_max(float v, float* sh) {
#pragma unroll
  for (int o = 16; o > 0; o >>= 1) v = fmaxf(v, __shfl_xor(v, o, 32));
  const int lane = threadIdx.x & 31, w = threadIdx.x >> 5;
  const int nw = (blockDim.x + 31) >> 5;
  __syncthreads();
  if (lane == 0) sh[w] = v;
  __syncthreads();
  if (w == 0) {
    float s = (lane < nw) ? sh[lane] : -3.402823466e38f;
#pragma unroll
    for (int o = 16; o > 0; o >>= 1) s = fmaxf(s, __shfl_xor(s, o, 32));
    if (lane == 0) sh[0] = s;
  }
  __syncthreads();
  return sh[0];
}

// ----------------------------- fp32 WMMA GEMM -----------------------------
// C[M,Ncols] = op( A[M,K] * Bmat )   (optionally +Add, optionally relu)
// BLAYOUT==0 : Bmat given as [Ncols,K] row-major (i.e. C = A * Bmat^T)
// BLAYOUT==1 : Bmat given as [K,Ncols] row-major (i.e. C = A * Bmat)
// Batched via blockIdx.z with per-operand batch strides (0 => shared).
#define GBM 64
#define GBN 128
#define GKC 32
#define GPITCH (GKC + 2)

template <int BLAYOUT, int RELU, int ADDC>
__global__ __launch_bounds__(256) void gemm_wmma(
    const float* __restrict__ A, int lda, long sAb,
    const float* __restrict__ Bg, int ldb, long sBb,
    float* __restrict__ C, int ldc, long sCb,
    const float* __restrict__ Add, long sDb,
    int M, int Ncols, int Kact) {
  __shared__ float As[GBM][GPITCH];
  __shared__ float Bs[GBN][GPITCH];

  const int t = threadIdx.x;
  const int lane = t & 31;
  const int wave = t >> 5;
  const long bz = blockIdx.z;
  const int m0 = blockIdx.y * GBM;
  const int n0 = blockIdx.x * GBN;

  A += bz * sAb;
  Bg += bz * sBb;
  C += bz * sCb;
  if (ADDC) Add += bz * sDb;

  const int mT = wave & 3;   // M sub-tile 0..3
  const int nG = wave >> 2;  // N group 0..1 (4 tiles of 16 each)
  const int mrow = lane & 15;
  const int koff = (lane >> 4) << 1;  // 0 or 2 (K split across lane halves)

  v8f acc[4] = {v8f{}, v8f{}, v8f{}, v8f{}};

  const int KT = (Kact + GKC - 1) / GKC;
  for (int kt = 0; kt < KT; ++kt) {
    const int kb = kt * GKC;
    // stage A tile: 64x32, 8 elems/thread, coalesced rows
#pragma unroll
    for (int i = 0; i < 8; ++i) {
      const int id = t + i * 256;
      const int r = id >> 5, c = id & 31;
      const int gr = m0 + r, gc = kb + c;
      float v = 0.f;
      if (gr < M && gc < Kact) v = A[(long)gr * lda + gc];
      As[r][c] = v;
    }
    // stage B tile in "Bt" layout: Bs[n][k]
    if (BLAYOUT == 0) {
#pragma unroll
      for (int i = 0; i < 16; ++i) {
        const int id = t + i * 256;
        const int r = id >> 5, c = id & 31;
        const int gn = n0 + r, gk = kb + c;
        float v = 0.f;
        if (gn < Ncols && gk < Kact) v = Bg[(long)gn * ldb + gk];
        Bs[r][c] = v;
      }
    } else {
#pragma unroll
      for (int i = 0; i < 16; ++i) {
        const int id = t + i * 256;
        const int n = id & 127, k = id >> 7;
        const int gn = n0 + n, gk = kb + k;
        float v = 0.f;
        if (gn < Ncols && gk < Kact) v = Bg[(long)gk * ldb + gn];
        Bs[n][k] = v;
      }
    }
    __syncthreads();

    const int aRow = mT * 16 + mrow;
#pragma unroll
    for (int kk = 0; kk < GKC; kk += 4) {
      v2f a;
      a.x = As[aRow][kk + koff];
      a.y = As[aRow][kk + koff + 1];
#pragma unroll
      for (int j = 0; j < 4; ++j) {
        const int bRow = (nG * 4 + j) * 16 + mrow;
        v2f b;
        b.x = Bs[bRow][kk + koff];
        b.y = Bs[bRow][kk + koff + 1];
        acc[j] = __builtin_amdgcn_wmma_f32_16x16x4_f32(
            false, a, false, b, (short)0, acc[j], false, false);
      }
    }
    __syncthreads();
  }

  // epilogue: C/D layout — VGPR v: rows v (lanes 0-15) and v+8 (lanes 16-31)
#pragma unroll
  for (int j = 0; j < 4; ++j) {
    const int col = n0 + (nG * 4 + j) * 16 + (lane & 15);
#pragma unroll
    for (int v = 0; v < 8; ++v) {
      const int row = m0 + mT * 16 + v + ((lane >> 4) << 3);
      if (row < M && col < Ncols) {
        float x = acc[j][v];
        if (ADDC) x += Add[(long)row * ldc + col];
        if (RELU) x = fmaxf(x, 0.f);
        C[(long)row * ldc + col] = x;
      }
    }
  }
}

// ----------------------------- rowwise kernels -----------------------------
// LayerNorm over rows of length 64 (fusion = ln(nodevec_fusion))
__global__ __launch_bounds__(64) void ln64_kernel(const float* __restrict__ in,
                                                  float* __restrict__ out) {
  __shared__ float sh[8];
  const long r = blockIdx.x;
  const int t = threadIdx.x;
  const float x = in[r * 64 + t];
  const float s = block_sum(x, sh);
  const float s2 = block_sum(x * x, sh);
  const float m = s * (1.f / 64.f);
  const float var = s2 * (1.f / 64.f) - m * m;
  out[r * 64 + t] = (x - m) * rsqrtf(var + LEPS);
}

// q/k projection: leaky_relu(fusion @ W), with head weights (1+colsum(mlp_w))/4 folded into q
__global__ __launch_bounds__(128) void qk_kernel(
    const float* __restrict__ F, const float* __restrict__ wq,
    const float* __restrict__ wk, const float* __restrict__ mlpw,
    float* __restrict__ Q, float* __restrict__ Kb) {
  __shared__ float fr[64];
  __shared__ float wp[8];
  const long r = blockIdx.x;
  const int t = threadIdx.x;
  if (t < 64) fr[t] = F[r * 64 + t];
  if (t < 8) {
    float s = 0.f;
    for (int o = 0; o < 8; ++o) s += mlpw[o * 8 + t];
    wp[t] = (1.f + s) * 0.25f;  // /sqrt(hd)=... hd=16 -> 1/4
  }
  __syncthreads();
  float aq = 0.f, ak = 0.f;
  for (int d = 0; d < 64; ++d) {
    const float f = fr[d];
    aq += f * wq[d * 128 + t];
    ak += f * wk[d * 128 + t];
  }
  aq = (aq > 0.f) ? aq : 0.01f * aq;  // leaky_relu slope 0.01
  ak = (ak > 0.f) ? ak : 0.01f * ak;
  Q[r * 128 + t] = aq * wp[t >> 4];
  Kb[r * 128 + t] = ak;
}

// adj_static: per row ln -> relu -> softmax ; store to Ab and 16x broadcast to output
__global__ __launch_bounds__(256) void static_kernel(
    const float* __restrict__ S, float* __restrict__ Ab,
    float* __restrict__ outStat) {
  __shared__ float sh[8];
  const int r = blockIdx.x;
  const int t = threadIdx.x;
  const float* row = S + (long)r * NPAD;
  float xv[4];
  float s = 0.f, s2 = 0.f;
#pragma unroll
  for (int i = 0; i < 4; ++i) {
    const int j = t + i * 256;
    xv[i] = (j < NNODES) ? row[j] : 0.f;
    s += xv[i];
    s2 += xv[i] * xv[i];
  }
  s = block_sum(s, sh);
  s2 = block_sum(s2, sh);
  const float inv = 1.f / (float)NNODES;
  const float m = s * inv;
  const float rs = rsqrtf(s2 * inv - m * m + LEPS);
  float mx = -3.402823466e38f;
#pragma unroll
  for (int i = 0; i < 4; ++i) {
    const int j = t + i * 256;
    if (j < NNODES) {
      xv[i] = fmaxf((xv[i] - m) * rs, 0.f);
      mx = fmaxf(mx, xv[i]);
    }
  }
  mx = block_max(mx, sh);
  float se = 0.f;
  float ev[4];
#pragma unroll
  for (int i = 0; i < 4; ++i) {
    const int j = t + i * 256;
    ev[i] = (j < NNODES) ? __expf(xv[i] - mx) : 0.f;
    se += ev[i];
  }
  se = block_sum(se, sh);
  const float isum = 1.f / se;
#pragma unroll
  for (int i = 0; i < 4; ++i) {
    const int j = t + i * 256;
    if (j < NNODES) {
      const float p = ev[i] * isum;
      Ab[(long)r * NPAD + j] = p;
      for (int b = 0; b < NB; ++b)
        outStat[((long)b * NNODES + r) * NNODES + j] = p;
    }
  }
}

// in-place layernorm of a [*, NPAD]-strided row of valid length 1000
__global__ __launch_bounds__(256) void rowln_kernel(float* __restrict__ Buf) {
  __shared__ float sh[8];
  const long r = blockIdx.x;
  float* row = Buf + r * NPAD;
  const int t = threadIdx.x;
  float xv[4];
  float s = 0.f, s2 = 0.f;
#pragma unroll
  for (int i = 0; i < 4; ++i) {
    const int j = t + i * 256;
    xv[i] = (j < NNODES) ? row[j] : 0.f;
    s += xv[i];
    s2 += xv[i] * xv[i];
  }
  s = block_sum(s, sh);
  s2 = block_sum(s2, sh);
  const float inv = 1.f / (float)NNODES;
  const float m = s * inv;
  const float rs = rsqrtf(s2 * inv - m * m + LEPS);
#pragma unroll
  for (int i = 0; i < 4; ++i) {
    const int j = t + i * 256;
    if (j < NNODES) row[j] = (xv[i] - m) * rs;
  }
}

// X <- adj_bf = ln(X)+ln(Y) ;  Y <- ln(adj_bf)
__global__ __launch_bounds__(256) void lnln_kernel(float* __restrict__ Xb,
                                                   float* __restrict__ Yb) {
  __shared__ float sh[8];
  const long r = blockIdx.x;
  float* xr = Xb + r * NPAD;
  float* yr = Yb + r * NPAD;
  const int t = threadIdx.x;
  float xv[4], yv[4];
  float sx = 0.f, sx2 = 0.f, sy = 0.f, sy2 = 0.f;
#pragma unroll
  for (int i = 0; i < 4; ++i) {
    const int j = t + i * 256;
    const bool ok = j < NNODES;
    xv[i] = ok ? xr[j] : 0.f;
    yv[i] = ok ? yr[j] : 0.f;
    sx += xv[i]; sx2 += xv[i] * xv[i];
    sy += yv[i]; sy2 += yv[i] * yv[i];
  }
  sx = block_sum(sx, sh); sx2 = block_sum(sx2, sh);
  sy = block_sum(sy, sh); sy2 = block_sum(sy2, sh);
  const float inv = 1.f / (float)NNODES;
  const float mx = sx * inv, rx = rsqrtf(sx2 * inv - mx * mx + LEPS);
  const float my = sy * inv, ry = rsqrtf(sy2 * inv - my * my + LEPS);
  float av[4];
  float sa = 0.f, sa2 = 0.f;
#pragma unroll
  for (int i = 0; i < 4; ++i) {
    const int j = t + i * 256;
    const bool ok = j < NNODES;
    av[i] = ok ? ((xv[i] - mx) * rx + (yv[i] - my) * ry) : 0.f;
    if (ok) xr[j] = av[i];
    sa += av[i]; sa2 += av[i] * av[i];
  }
  sa = block_sum(sa, sh); sa2 = block_sum(sa2, sh);
  const float ma = sa * inv, ra = rsqrtf(sa2 * inv - ma * ma + LEPS);
#pragma unroll
  for (int i = 0; i < 4; ++i) {
    const int j = t + i * 256;
    if (j < NNODES) yr[j] = (av[i] - ma) * ra;
  }
}

// final: out = softmax(relu(ln( ln(Y) + SG )), rowwise)
__global__ __launch_bounds__(256) void final_kernel(
    const float* __restrict__ Yb, const float* __restrict__ SG,
    float* __restrict__ outDyn) {
  __shared__ float sh[8];
  const int r = blockIdx.x, b = blockIdx.y;
  const float* yr = Yb + ((long)b * NNODES + r) * NPAD;
  const float* sg = SG + (long)r * NPAD;
  const int t = threadIdx.x;
  float yv[4], gv[4];
  float s = 0.f, s2 = 0.f;
#pragma unroll
  for (int i = 0; i < 4; ++i) {
    const int j = t + i * 256;
    const bool ok = j < NNODES;
    yv[i] = ok ? yr[j] : 0.f;
    gv[i] = ok ? sg[j] : 0.f;
    s += yv[i]; s2 += yv[i] * yv[i];
  }
  s = block_sum(s, sh); s2 = block_sum(s2, sh);
  const float inv = 1.f / (float)NNODES;
  float m = s * inv;
  float rs = rsqrtf(s2 * inv - m * m + LEPS);
  float uv[4];
  float su = 0.f, su2 = 0.f;
#pragma unroll
  for (int i = 0; i < 4; ++i) {
    const int j = t + i * 256;
    const bool ok = j < NNODES;
    uv[i] = ok ? ((yv[i] - m) * rs + gv[i]) : 0.f;
    su += uv[i]; su2 += uv[i] * uv[i];
  }
  su = block_sum(su, sh); su2 = block_sum(su2, sh);
  m = su * inv;
  rs = rsqrtf(su2 * inv - m * m + LEPS);
  float mx = -3.402823466e38f;
#pragma unroll
  for (int i = 0; i < 4; ++i) {
    const int j = t + i * 256;
    if (j < NNODES) {
      uv[i] = fmaxf((uv[i] - m) * rs, 0.f);
      mx = fmaxf(mx, uv[i]);
    }
  }
  mx = block_max(mx, sh);
  float se = 0.f;
  float ev[4];
#pragma unroll
  for (int i = 0; i < 4; ++i) {
    const int j = t + i * 256;
    ev[i] = (j < NNODES) ? __expf(uv[i] - mx) : 0.f;
    se += ev[i];
  }
  se = block_sum(se, sh);
  const float isum = 1.f / se;
#pragma unroll
  for (int i = 0; i < 4; ++i) {
    const int j = t + i * 256;
    if (j < NNODES) outDyn[((long)b * NNODES + r) * NNODES + j] = ev[i] * isum;
  }
}

// column sums + squared column sums of adj_static
__global__ void colsum_kernel(const float* __restrict__ Ab,
                              float* __restrict__ cs, float* __restrict__ fp) {
  const int j = blockIdx.x * 256 + threadIdx.x;
  if (j >= NNODES) return;
  float s = 0.f, f = 0.f;
  for (int i = 0; i < NNODES; ++i) {
    const float v = Ab[(long)i * NPAD + j];
    s += v;
    f += v * v;
  }
  cs[j] = s;
  fp[j] = f;
}

// gl_loss[b] = 1e-3 * ( sum x^2*(1+colsum[n]) - 2*sum(x .* (A@x)) ) + 1e-3 * ||A||_F^2
__global__ __launch_bounds__(256) void gl_kernel(
    const float* __restrict__ Xin, const float* __restrict__ AX,
    const float* __restrict__ cs, const float* __restrict__ fp,
    float* __restrict__ outGl) {
  __shared__ float sh[8];
  const int b = blockIdx.x;
  const int t = threadIdx.x;
  const float* x = Xin + (long)b * NNODES * DNODE;
  const float* ax = AX + (long)b * NNODES * DNODE;
  float acc = 0.f;
  for (int idx = t; idx < NNODES * DNODE; idx += 256) {
    const int n = idx >> 6;
    const float xv = x[idx];
    acc += xv * xv * (1.f + cs[n]) - 2.f * xv * ax[idx];
  }
  float fn = 0.f;
  for (int j = t; j < NNODES; j += 256) fn += fp[j];
  const float dist = block_sum(acc, sh);
  const float fns = block_sum(fn, sh);
  if (t == 0) outGl[b] = dist * 1e-3f + fns * 1e-3f;
}

__global__ void copy_kernel(const float* __restrict__ src,
                            float* __restrict__ dst, int n) {
  const int i = blockIdx.x * 256 + threadIdx.x;
  if (i < n) dst[i] = src[i];
}

// ----------------------------- launch -----------------------------
extern "C" void kernel_launch(void* const* d_in, const int* in_sizes, int n_in,
                              void* d_out, int out_size, void* d_ws,
                              size_t ws_size, hipStream_t stream) {
  (void)in_sizes; (void)n_in; (void)out_size; (void)ws_size;
  const float* nodevec_fusion = (const float*)d_in[0];
  const float* nodevec_s = (const float*)d_in[1];
  const float* node_input = (const float*)d_in[2];
  const float* nodevec_dy = (const float*)d_in[3];
  const float* w_q = (const float*)d_in[4];
  const float* w_k = (const float*)d_in[5];
  // d_in[6] node_w: dead — rowsum(ln(x)) == 0 exactly, so nv11 == fusion
  const float* mlp_w = (const float*)d_in[7];
  // d_in[8] mlp_b: dead — constant shift is absorbed by the following LayerNorm
  const float* attn_linear = (const float*)d_in[9];
  const float* attn_linear_1 = (const float*)d_in[10];

  float* out = (float*)d_out;
  float* o_dyn = out;                                   // [16,1000,1000]
  float* o_stat = out + (size_t)NB * NNODES * NNODES;   // [16,1000,1000]
  float* o_ns = out + 2ull * NB * NNODES * NNODES;      // [1000,64]
  float* o_gl = o_ns + (size_t)NNODES * DNODE;          // [16]

  float* ws = (float*)d_ws;
  size_t off = 0;
  float* F = ws + off;  off += (size_t)NB * NNODES * DNODE;   // ln(fusion)
  float* Q = ws + off;  off += (size_t)NB * NNODES * DMODEL;  // scaled q
  float* Kb = ws + off; off += (size_t)NB * NNODES * DMODEL;  // k
  float* X = ws + off;  off += (size_t)NB * NNODES * NPAD;    // adj_sum -> adj_bf
  float* Y = ws + off;  off += (size_t)NB * NNODES * NPAD;    // gram -> ln(adj_bf) -> adj_af+adj_bf
  float* Z = ws + off;  off += (size_t)NB * NNODES * NPAD;    // relu(ln(adj_bf)@W1)
  float* Sb = ws + off; off += (size_t)NNODES * NPAD;         // s gram
  float* Db = ws + off; off += (size_t)NNODES * NPAD;         // dy gram -> static_graph_inf
  float* Ab = ws + off; off += (size_t)NNODES * NPAD;         // adj_static
  float* AX = ws + off; off += (size_t)NB * NNODES * DNODE;   // adj_static @ node_input
  float* CS = ws + off; off += NPAD;                          // column sums
  float* FP = ws + off; off += NPAD;                          // column sq sums

  const long sF = (long)NNODES * DNODE;
  const long sQ = (long)NNODES * DMODEL;
  const long sN = (long)NNODES * NPAD;
  const dim3 gNN(8, 16, 1);    // [1000x1000] tiles
  const dim3 gBNN(8, 16, NB);  // batched [1000x1000]

  // 1) fusion = ln(nodevec_fusion)
  ln64_kernel<<<NB * NNODES, 64, 0, stream>>>(nodevec_fusion, F);
  // 2) Sb = s @ s^T ; Db = dy @ dy^T
  gemm_wmma<0, 0, 0><<<gNN, 256, 0, stream>>>(nodevec_s, DNODE, 0, nodevec_s,
      DNODE, 0, Sb, NPAD, 0, nullptr, 0, NNODES, NNODES, DNODE);
  gemm_wmma<0, 0, 0><<<gNN, 256, 0, stream>>>(nodevec_dy, DNODE, 0, nodevec_dy,
      DNODE, 0, Db, NPAD, 0, nullptr, 0, NNODES, NNODES, DNODE);
  // 3) adj_static = softmax(relu(ln(Sb))) ; broadcast to output
  static_kernel<<<NNODES, 256, 0, stream>>>(Sb, Ab, o_stat);
  // 4) static_graph_inf = ln(Db) in place
  rowln_kernel<<<NNODES, 256, 0, stream>>>(Db);
  // 5) Y = fusion @ fusion^T (nv11 gram)
  gemm_wmma<0, 0, 0><<<gBNN, 256, 0, stream>>>(F, DNODE, sF, F, DNODE, sF, Y,
      NPAD, sN, nullptr, 0, NNODES, NNODES, DNODE);
  // 6) q (scaled) and k projections
  qk_kernel<<<NB * NNODES, 128, 0, stream>>>(F, w_q, w_k, mlp_w, Q, Kb);
  // 7) X = q_s @ k^T  (head-summed attention, K=128)
  gemm_wmma<0, 0, 0><<<gBNN, 256, 0, stream>>>(Q, DMODEL, sQ, Kb, DMODEL, sQ, X,
      NPAD, sN, nullptr, 0, NNODES, NNODES, DMODEL);
  // 8) X = adj_bf = ln(X)+ln(Y) ; Y = ln(adj_bf)
  lnln_kernel<<<NB * NNODES, 256, 0, stream>>>(X, Y);
  // 9) Z = relu(Y @ attn_linear)
  gemm_wmma<1, 1, 0><<<gBNN, 256, 0, stream>>>(Y, NPAD, sN, attn_linear, NNODES,
      0, Z, NPAD, sN, nullptr, 0, NNODES, NNODES, NNODES);
  // 10) Y = Z @ attn_linear_1 + adj_bf
  gemm_wmma<1, 0, 1><<<gBNN, 256, 0, stream>>>(Z, NPAD, sN, attn_linear_1,
      NNODES, 0, Y, NPAD, sN, X, sN, NNODES, NNODES, NNODES);
  // 11) adj_dynamic = softmax(relu(ln(ln(Y)+static_graph_inf)))
  final_kernel<<<dim3(NNODES, NB), 256, 0, stream>>>(Y, Db, o_dyn);
  // 12) AX = adj_static @ node_input (per batch)
  gemm_wmma<1, 0, 0><<<dim3(1, 16, NB), 256, 0, stream>>>(Ab, NPAD, 0,
      node_input, DNODE, (long)NNODES * DNODE, AX, DNODE, (long)NNODES * DNODE,
      nullptr, 0, NNODES, DNODE, NNODES);
  // 13) column sums / Frobenius parts, then graph loss
  colsum_kernel<<<4, 256, 0, stream>>>(Ab, CS, FP);
  gl_kernel<<<NB, 256, 0, stream>>>(node_input, AX, CS, FP, o_gl);
  // 14) passthrough nodevec_s
  copy_kernel<<<(NNODES * DNODE + 255) / 256, 256, 0, stream>>>(
      nodevec_s, o_ns, NNODES * DNODE);
}